// NIQE_45397804318933
// MI455X (gfx1250) — compile-verified
//
#include <hip/hip_runtime.h>
#include <hip/hip_bf16.h>
#include <math.h>

// ---------------------------------------------------------------------------
// NIQE-style metric, MI455X (gfx1250).
// Stage 1: per-48x48-cell sufficient statistics (bandwidth bound, one pass,
//          async global->LDS staging).
// Stage 2: combine 2x2 cells (+crossing terms) -> per-patch stats -> feats.
// Stage 3: 12x12 Gauss-Jordan inverse + Mahalanobis GEMM via f32 WMMA.
// ---------------------------------------------------------------------------

#define HW      1056
#define STRIDEP 48
#define NCELL   22                 // 1056 / 48
#define NPATCH  21                 // (1056-96)/48 + 1
#define TILE_W  50                 // padded LDS row stride (floats), 49 cols used
#define CELLS_PER_IMG (NCELL*NCELL)  // 484
#define NIMG    32                 // B*C = 8*4

typedef float v2f __attribute__((ext_vector_type(2)));
typedef float v8f __attribute__((ext_vector_type(8)));

__device__ __forceinline__ float wave_reduce_add(float v) {
  #pragma unroll
  for (int off = 16; off > 0; off >>= 1)
    v += __shfl_down(v, off, 32);
  return v;
}

// ---------------------------------------------------------------------------
// Kernel 1: one block per (image, cell). Stage 49x49 tile into LDS using the
// CDNA5 async global->LDS path, then compute 6 cell statistics:
//   [0]=sum x, [1]=sum x^2, [2]=interior gx^2, [3]=interior gy^2,
//   [4]=right-boundary-crossing gx^2, [5]=bottom-boundary-crossing gy^2
// ---------------------------------------------------------------------------
__global__ __launch_bounds__(256) void cell_stats_kernel(
    const float* __restrict__ x, float* __restrict__ cells)
{
  __shared__ float tile[49 * TILE_W];
  __shared__ float acc[6];

  const int t    = threadIdx.x;
  const int blk  = blockIdx.x;
  const int cell = blk % CELLS_PER_IMG;
  const int img  = blk / CELLS_PER_IMG;
  const int ci   = cell / NCELL;
  const int cj   = cell % NCELL;
  const int r0   = ci * STRIDEP;
  const int c0   = cj * STRIDEP;
  const float* base = x + (size_t)img * (size_t)(HW * HW);

  if (t < 6) acc[t] = 0.0f;

  // Low 32 bits of a generic pointer to LDS are the LDS byte offset.
  const unsigned lds_base = (unsigned)(unsigned long long)(const void*)tile;

  // Async-stage main 49 rows x 48 cols: 12 x B128 per row (16B-aligned:
  // c0 = 48*cj -> 192*cj bytes; row pitch 1056*4 is 16B-aligned).
  for (int chunk = t; chunk < 49 * 12; chunk += 256) {
    int row  = chunk / 12;
    int quad = chunk % 12;
    int gr   = r0 + row; if (gr > HW - 1) gr = HW - 1;   // clamp (edge rows unused)
    unsigned goff = (unsigned)(((gr * HW) + c0 + quad * 4) * 4);
    unsigned loff = lds_base + (unsigned)((row * TILE_W + quad * 4) * 4);
    asm volatile("global_load_async_to_lds_b128 %0, %1, %2"
                 :: "v"(loff), "v"(goff), "s"(base)
                 : "memory");
  }
  // Extra right-neighbor column (col 48) for 49 rows, plain loads (distinct
  // LDS addresses from the async region).
  if (t < 49) {
    int gr = r0 + t;  if (gr > HW - 1) gr = HW - 1;
    int gc = c0 + 48; if (gc > HW - 1) gc = HW - 1;      // clamp (edge col unused)
    tile[t * TILE_W + 48] = base[(size_t)gr * HW + gc];
  }
#if __has_builtin(__builtin_amdgcn_s_wait_asynccnt)
  __builtin_amdgcn_s_wait_asynccnt(0);
#else
  asm volatile("s_wait_asynccnt 0" ::: "memory");
#endif
  __syncthreads();

  float S = 0.f, Q = 0.f, GX = 0.f, GY = 0.f, CX = 0.f, CY = 0.f;
  for (int idx = t; idx < 48 * 48; idx += 256) {
    int r  = idx / 48;
    int cc = idx % 48;
    float v  = tile[r * TILE_W + cc];
    float vr = tile[r * TILE_W + cc + 1];
    float vd = tile[(r + 1) * TILE_W + cc];
    S += v;
    Q += v * v;
    float dx = vr - v;
    float dy = vd - v;
    if (cc < 47) GX += dx * dx; else CX += dx * dx;
    if (r  < 47) GY += dy * dy; else CY += dy * dy;
  }

  S  = wave_reduce_add(S);
  Q  = wave_reduce_add(Q);
  GX = wave_reduce_add(GX);
  GY = wave_reduce_add(GY);
  CX = wave_reduce_add(CX);
  CY = wave_reduce_add(CY);
  if ((t & 31) == 0) {
    atomicAdd(&acc[0], S);
    atomicAdd(&acc[1], Q);
    atomicAdd(&acc[2], GX);
    atomicAdd(&acc[3], GY);
    atomicAdd(&acc[4], CX);
    atomicAdd(&acc[5], CY);
  }
  __syncthreads();
  if (t == 0) {
    float* o = cells + (size_t)(img * CELLS_PER_IMG + cell) * 6;
    o[0] = acc[0]; o[1] = acc[1]; o[2] = acc[2];
    o[3] = acc[3]; o[4] = acc[4]; o[5] = acc[5];
  }
}

// ---------------------------------------------------------------------------
// Kernel 2: one block per image (b,c). Combine 2x2 cells per patch, compute
// per-patch mu / var(ddof=1) / grad_mag, average over 441 patches -> feats.
// feats layout: feats[b*12 + f], f = c (mu), 4+c (var), 8+c (grad).
// ---------------------------------------------------------------------------
__global__ __launch_bounds__(256) void patch_feats_kernel(
    const float* __restrict__ cells, float* __restrict__ feats)
{
  __shared__ float acc[3];
  const int t   = threadIdx.x;
  const int img = blockIdx.x;           // 0..31
  const float* cs = cells + (size_t)img * CELLS_PER_IMG * 6;

  if (t < 3) acc[t] = 0.0f;
  __syncthreads();

  float muA = 0.f, vaA = 0.f, grA = 0.f;
  for (int p = t; p < NPATCH * NPATCH; p += 256) {
    int ph = p / NPATCH, pw = p % NPATCH;
    const float* c00 = cs + ((ph    ) * NCELL + pw    ) * 6;
    const float* c01 = cs + ((ph    ) * NCELL + pw + 1) * 6;
    const float* c10 = cs + ((ph + 1) * NCELL + pw    ) * 6;
    const float* c11 = cs + ((ph + 1) * NCELL + pw + 1) * 6;
    float S  = c00[0] + c01[0] + c10[0] + c11[0];
    float Q  = c00[1] + c01[1] + c10[1] + c11[1];
    float GX = c00[2] + c01[2] + c10[2] + c11[2] + c00[4] + c10[4];
    float GY = c00[3] + c01[3] + c10[3] + c11[3] + c00[5] + c01[5];
    const float N = 9216.0f;            // 96*96
    float mu  = S / N;
    float var = (Q - S * S / N) / (N - 1.0f);
    float gm  = sqrtf(GX * (1.0f / 9120.0f) + GY * (1.0f / 9120.0f)); // 96*95
    muA += mu; vaA += var; grA += gm;
  }
  muA = wave_reduce_add(muA);
  vaA = wave_reduce_add(vaA);
  grA = wave_reduce_add(grA);
  if ((t & 31) == 0) {
    atomicAdd(&acc[0], muA);
    atomicAdd(&acc[1], vaA);
    atomicAdd(&acc[2], grA);
  }
  __syncthreads();
  if (t == 0) {
    const float inv_np = 1.0f / (NPATCH * NPATCH);
    int b = img / 4, c = img % 4;
    feats[b * 12 + c]     = acc[0] * inv_np;
    feats[b * 12 + 4 + c] = acc[1] * inv_np;
    feats[b * 12 + 8 + c] = acc[2] * inv_np;
  }
}

// ---------------------------------------------------------------------------
// Kernel 3: single wave (32 lanes, EXEC all-ones at the WMMA).
//   1) Gauss-Jordan inverse of cov[:12,:12] (thread 0, trivial size).
//   2) diff = feats - mu0, zero-padded to 16x16.
//   3) M = diff @ cov_inv via V_WMMA_F32_16X16X4_F32 (K = 12 = 3 k-steps).
//   4) dist_b = sqrt(sum_n M[b,n]*diff[b,n]); out = mean over 8 batches.
// ---------------------------------------------------------------------------
__global__ __launch_bounds__(32) void mahal_kernel(
    const float* __restrict__ feats, const float* __restrict__ mu_pr,
    const float* __restrict__ cov_pr, float* __restrict__ out)
{
  __shared__ float aug[12][24];
  __shared__ float invS[12][16];
  __shared__ float Dm[16][16];
  __shared__ float Mm[16][16];
  const int lane = threadIdx.x;

  for (int i = lane; i < 16 * 16; i += 32) ((float*)Dm)[i] = 0.0f;
  for (int i = lane; i < 12 * 16; i += 32) ((float*)invS)[i] = 0.0f;
  __syncthreads();
  for (int i = lane; i < 8 * 12; i += 32) {
    int b = i / 12, f = i % 12;
    Dm[b][f] = feats[b * 12 + f] - mu_pr[f];
  }
  __syncthreads();

  if (lane == 0) {
    for (int r = 0; r < 12; ++r)
      for (int cl = 0; cl < 12; ++cl) {
        aug[r][cl]      = cov_pr[r * 36 + cl];
        aug[r][12 + cl] = (r == cl) ? 1.0f : 0.0f;
      }
    for (int col = 0; col < 12; ++col) {
      int piv = col; float best = fabsf(aug[col][col]);
      for (int r = col + 1; r < 12; ++r) {
        float a = fabsf(aug[r][col]);
        if (a > best) { best = a; piv = r; }
      }
      if (piv != col)
        for (int cl = 0; cl < 24; ++cl) {
          float tmp = aug[col][cl]; aug[col][cl] = aug[piv][cl]; aug[piv][cl] = tmp;
        }
      float inv_d = 1.0f / aug[col][col];
      for (int cl = 0; cl < 24; ++cl) aug[col][cl] *= inv_d;
      for (int r = 0; r < 12; ++r)
        if (r != col) {
          float f = aug[r][col];
          if (f != 0.0f)
            for (int cl = 0; cl < 24; ++cl) aug[r][cl] -= f * aug[col][cl];
        }
    }
    for (int r = 0; r < 12; ++r)
      for (int n = 0; n < 12; ++n) invS[r][n] = aug[r][12 + n];
  }
  __syncthreads();

  const int half = lane >> 4;   // 0: K=0/1 slice, 1: K=2/3 slice (per ISA layout)
  const int lid  = lane & 15;
  v8f c = {0.f, 0.f, 0.f, 0.f, 0.f, 0.f, 0.f, 0.f};
#if __has_builtin(__builtin_amdgcn_wmma_f32_16x16x4_f32)
  #pragma unroll
  for (int kb = 0; kb < 3; ++kb) {
    v2f a, b;
    // A 16x4 f32: VGPR0 = K=0 (lanes 0-15) / K=2 (lanes 16-31); VGPR1 = K=1 / K=3
    a.x = Dm[lid][kb * 4 + 2 * half + 0];
    a.y = Dm[lid][kb * 4 + 2 * half + 1];
    // B 4x16 f32: VGPR0 = row K=0 (lanes 0-15) / K=2 (lanes 16-31); VGPR1 = K=1 / K=3
    b.x = invS[kb * 4 + 2 * half + 0][lid];
    b.y = invS[kb * 4 + 2 * half + 1][lid];
    c = __builtin_amdgcn_wmma_f32_16x16x4_f32(false, a, false, b, (short)0, c,
                                              false, false);
  }
  #pragma unroll
  for (int v = 0; v < 8; ++v)          // D: VGPR v = row v (lanes 0-15) / v+8
    Mm[v + 8 * half][lid] = c[v];
#else
  for (int i = lane; i < 16 * 16; i += 32) {
    int m = i / 16, n = i % 16;
    float s = 0.f;
    for (int k = 0; k < 12; ++k) s += Dm[m][k] * invS[k][n];
    Mm[m][n] = s;
  }
#endif
  __syncthreads();

  if (lane == 0) {
    float s = 0.f;
    for (int b = 0; b < 8; ++b) {
      float d2 = 0.f;
      for (int n = 0; n < 12; ++n) d2 += Mm[b][n] * Dm[b][n];
      s += sqrtf(fmaxf(d2, 0.0f));
    }
    out[0] = s * (1.0f / 8.0f);
  }
}

// ---------------------------------------------------------------------------
extern "C" void kernel_launch(void* const* d_in, const int* in_sizes, int n_in,
                              void* d_out, int out_size, void* d_ws, size_t ws_size,
                              hipStream_t stream) {
  (void)in_sizes; (void)n_in; (void)out_size; (void)ws_size;
  const float* x      = (const float*)d_in[0];
  const float* mu_pr  = (const float*)d_in[1];
  const float* cov_pr = (const float*)d_in[2];

  float* cells = (float*)d_ws;                                   // 32*484*6 f32
  float* feats = (float*)((char*)d_ws +
                          (size_t)NIMG * CELLS_PER_IMG * 6 * sizeof(float));

  cell_stats_kernel<<<NIMG * CELLS_PER_IMG, 256, 0, stream>>>(x, cells);
  patch_feats_kernel<<<NIMG, 256, 0, stream>>>(cells, feats);
  mahal_kernel<<<1, 32, 0, stream>>>(feats, mu_pr, cov_pr, (float*)d_out);
}